// custom_loss_1297080123505
// MI455X (gfx1250) — compile-verified
//
#include <hip/hip_runtime.h>
#include <hip/hip_bf16.h>
#include <stdint.h>

#define TPB    256
#define CELL_F 30
#define NCLS   20

#if __has_builtin(__builtin_amdgcn_global_load_async_to_lds_b128)
#define ATHENA_ASYNC 1
#else
#define ATHENA_ASYNC 0
#endif

#if ATHENA_ASYNC
typedef int athena_v4i __attribute__((ext_vector_type(4)));

__device__ __forceinline__ void athena_async_b128(const void* g, void* l) {
  // Signature (from clang diagnostic): (v4i AS1*, v4i AS3*, imm offset, imm cpol)
  __builtin_amdgcn_global_load_async_to_lds_b128(
      (__attribute__((address_space(1))) athena_v4i*)(g),
      (__attribute__((address_space(3))) athena_v4i*)(l), 0, 0);
}
#endif

__device__ __forceinline__ void athena_wait_async0() {
#if __has_builtin(__builtin_amdgcn_s_wait_asynccnt)
  __builtin_amdgcn_s_wait_asynccnt(0);
#else
  asm volatile("s_wait_asynccnt 0" ::: "memory");
#endif
}

__global__ __launch_bounds__(TPB) void yolo_zero_ws(float* __restrict__ ws) {
  if (threadIdx.x < 16) ws[threadIdx.x] = 0.0f;
}

// ws[0..7]: n_obj, contain_num, obj_num, notcontain_num, noobj_num, class_num,
//           sum_min_iou, acc_count
__global__ __launch_bounds__(TPB) void yolo_loss_partial(
    const float* __restrict__ outp, const float* __restrict__ tgtp,
    float* __restrict__ ws, int n_cells) {
  __shared__ float sOut[TPB * CELL_F];   // 30720 B
  __shared__ float sTgt[TPB * CELL_F];   // 30720 B
  __shared__ float sRed[(TPB / 32) * 8]; // cross-wave partials

  const int tid = threadIdx.x;
  const long long tileCell = (long long)blockIdx.x * TPB;
  const long long base = tileCell * CELL_F;
  const long long rem = (long long)n_cells - tileCell;
  const int cellsHere = (rem < (long long)TPB) ? (int)rem : TPB;

#if ATHENA_ASYNC
  {
    // Coalesced async global->LDS staging: 16B per lane per issue (ASYNCcnt path)
    const char* gO = (const char*)(outp + base);
    const char* gT = (const char*)(tgtp + base);
    char* lO = (char*)sOut;
    char* lT = (char*)sTgt;
    const int nChunks = (cellsHere * CELL_F * 4 + 15) >> 4;  // 1920 for full tile
#pragma unroll
    for (int j = 0; j < (TPB * CELL_F * 4 / 16 + TPB - 1) / TPB; ++j) {
      const int chunk = tid + j * TPB;
      if (chunk < nChunks) {
        athena_async_b128(gO + (size_t)chunk * 16, lO + (size_t)chunk * 16);
        athena_async_b128(gT + (size_t)chunk * 16, lT + (size_t)chunk * 16);
      }
    }
    athena_wait_async0();
  }
#else
  {
    const int nF = cellsHere * CELL_F;
    for (int j = tid; j < nF; j += TPB) {
      sOut[j] = outp[base + j];
      sTgt[j] = tgtp[base + j];
    }
  }
#endif
  __syncthreads();

  float vals[8];
#pragma unroll
  for (int j = 0; j < 8; ++j) vals[j] = 0.0f;

  if (tid < cellsHere) {
    float o[CELL_F], t[CELL_F];
#pragma unroll
    for (int k = 0; k < CELL_F; ++k) {
      o[k] = sOut[tid * CELL_F + k];   // stride 30 words: conflict-free in wave32
      t[k] = sTgt[tid * CELL_F + k];
    }
    const float m = (t[4] > 0.0f) ? 1.0f : 0.0f;

    // target corners + area
    const float tx0 = t[0] / 7.0f - 0.5f * t[2];
    const float ty0 = t[1] / 7.0f - 0.5f * t[3];
    const float tx1 = t[0] / 7.0f + 0.5f * t[2];
    const float ty1 = t[1] / 7.0f + 0.5f * t[3];
    const float at = (tx1 - tx0) * (ty1 - ty0);

    float iou[2];
#pragma unroll
    for (int b = 0; b < 2; ++b) {
      const float px0 = o[5 * b + 0] / 7.0f - 0.5f * o[5 * b + 2];
      const float py0 = o[5 * b + 1] / 7.0f - 0.5f * o[5 * b + 3];
      const float px1 = o[5 * b + 0] / 7.0f + 0.5f * o[5 * b + 2];
      const float py1 = o[5 * b + 1] / 7.0f + 0.5f * o[5 * b + 3];
      const float ap = (px1 - px0) * (py1 - py0);
      const float wi = fmaxf(fminf(px1, tx1) - fmaxf(px0, tx0), 0.0f);
      const float hi = fmaxf(fminf(py1, ty1) - fmaxf(py0, ty0), 0.0f);
      const float inter = wi * hi;
      iou[b] = inter / (ap + at - inter);
    }
    // argmax-first semantics: tie -> box 0
    const int resp = (iou[1] > iou[0]) ? 1 : 0;
    const float max_iou = resp ? iou[1] : iou[0];
    const float min_iou = resp ? iou[0] : iou[1];
    const int rb = 5 * resp;
    const int nb = 5 * (1 - resp);

    const float c0 = o[rb + 0] - t[0];
    const float c1 = o[rb + 1] - t[1];
    const float c2 = sqrtf(o[rb + 2]) - sqrtf(t[2]);
    const float c3 = sqrtf(o[rb + 3]) - sqrtf(t[3]);
    const float dresp = o[rb + 4] - max_iou;
    const float d4 = o[4] - t[4];
    const float d9 = o[9] - t[9];

    float cls = 0.0f;
    int pa = 0, ta = 0;
    float pv = o[10], tv = t[10];
#pragma unroll
    for (int c = 0; c < NCLS; ++c) {
      const float dc = o[10 + c] - t[10 + c];
      cls += dc * dc;
      if (c > 0) {  // first-max argmax, matching jnp.argmax
        if (o[10 + c] > pv) { pv = o[10 + c]; pa = c; }
        if (t[10 + c] > tv) { tv = t[10 + c]; ta = c; }
      }
    }

    vals[0] = m;
    vals[1] = m * (c0 * c0 + c1 * c1 + c2 * c2 + c3 * c3);
    vals[2] = m * dresp * dresp;
    vals[3] = m * o[nb + 4] * o[nb + 4];
    vals[4] = (1.0f - m) * (d4 * d4 + d9 * d9);
    vals[5] = m * cls;
    vals[6] = m * min_iou;
    vals[7] = (m > 0.0f && pa == ta) ? 1.0f : 0.0f;
  }

  // wave32 shuffle tree -> per-wave leader -> LDS -> one atomic set per block
#pragma unroll
  for (int j = 0; j < 8; ++j) {
    float v = vals[j];
#pragma unroll
    for (int off = 16; off > 0; off >>= 1) v += __shfl_down(v, off, 32);
    if ((tid & 31) == 0) sRed[(tid >> 5) * 8 + j] = v;
  }
  __syncthreads();
  if (tid == 0) {
#pragma unroll
    for (int j = 0; j < 8; ++j) {
      float s = 0.0f;
#pragma unroll
      for (int w = 0; w < TPB / 32; ++w) s += sRed[w * 8 + j];
      atomicAdd(&ws[j], s);
    }
  }
}

__global__ __launch_bounds__(32) void yolo_finalize(const float* __restrict__ ws,
                                                    float* __restrict__ outv,
                                                    int n_cells) {
  if (threadIdx.x == 0 && blockIdx.x == 0) {
    const float n_obj = ws[0];
    const float n_noobj = (float)n_cells - n_obj;
    const float contain = ws[1] / (2.0f * n_obj);
    const float obj_loss = ws[2] / n_obj;
    const float not_contain = ws[3] / n_obj;
    const float noobj_loss = ws[4] / (2.0f * n_noobj);
    const float class_loss = ws[5] / (n_obj * (float)NCLS);
    outv[0] = 5.0f * contain + obj_loss + 0.5f * (noobj_loss + not_contain) + class_loss;
    outv[1] = ws[6];  // sum_iou
    outv[2] = ws[7];  // acc (count as float)
  }
}

extern "C" void kernel_launch(void* const* d_in, const int* in_sizes, int n_in,
                              void* d_out, int out_size, void* d_ws, size_t ws_size,
                              hipStream_t stream) {
  const float* outp = (const float*)d_in[0];
  const float* tgtp = (const float*)d_in[1];
  float* ws = (float*)d_ws;
  float* res = (float*)d_out;

  const int n_cells = in_sizes[0] / CELL_F;  // 8192*7*7 = 401408
  const int blocks = (n_cells + TPB - 1) / TPB;

  yolo_zero_ws<<<1, 32, 0, stream>>>(ws);
  yolo_loss_partial<<<blocks, TPB, 0, stream>>>(outp, tgtp, ws, n_cells);
  yolo_finalize<<<1, 32, 0, stream>>>(ws, res, n_cells);
}